// DACRVQVAEBottleneck_23957327577862
// MI455X (gfx1250) — compile-verified
//
#include <hip/hip_runtime.h>
#include <hip/hip_bf16.h>
#include <math.h>

typedef __attribute__((ext_vector_type(2))) float v2f;
typedef __attribute__((ext_vector_type(8))) float v8f;

#define NCB  9
#define NBAT 8
#define ND   1024
#define NT   4096
#define NDCB 8
#define NK   1024
#define TT   16      // time-steps per workgroup
#define RS   18      // residual LDS row stride (floats) -> conflict-free
#define ZS   17      // small tile row stride

// ---------------------------------------------------------------------------
// Pre-kernel 1: L2-normalize codebooks, store transposed AND row-pair
// interleaved:  cbnP[i][pair][k][2]  (pair = d>>1, slot = d&1), so each
// search-GEMM B-fragment is a single aligned global_load_b64.
// ---------------------------------------------------------------------------
__global__ __launch_bounds__(256) void cbnorm_kernel(const float* __restrict__ cb,
                                                     float* __restrict__ cbnP) {
  int g = blockIdx.x * blockDim.x + threadIdx.x;   // g = i*1024 + k
  if (g >= NCB * NK) return;
  int i = g >> 10;
  int k = g & (NK - 1);
  const float* v = cb + (size_t)g * NDCB;
  float s = 0.f;
#pragma unroll
  for (int d = 0; d < NDCB; ++d) s += v[d] * v[d];
  float inv = 1.f / fmaxf(sqrtf(s), 1e-12f);
#pragma unroll
  for (int d = 0; d < NDCB; ++d)
    cbnP[((size_t)(i * 4 + (d >> 1)) * NK + k) * 2 + (d & 1)] = v[d] * inv;
}

// ---------------------------------------------------------------------------
// Pre-kernel 2: zero-pad W_in rows 8..15 -> Wpad[i][16][1024] so the in-proj
// A-fragment load is UNCONDITIONAL (no per-chunk EXEC masking / divergence).
// ---------------------------------------------------------------------------
__global__ __launch_bounds__(256) void wpad_kernel(const float* __restrict__ W_in,
                                                   float* __restrict__ Wpad) {
  int g = blockIdx.x * blockDim.x + threadIdx.x;   // g = ((i*16)+row)*1024 + k
  if (g >= NCB * 16 * ND) return;
  int k   = g & (ND - 1);
  int row = (g >> 10) & 15;
  int i   = g >> 14;
  float v = 0.f;
  if (row < NDCB) v = W_in[((size_t)i * NDCB + row) * ND + k];
  Wpad[g] = v;
}

__device__ __forceinline__ v8f wmma_f32(v2f a, v2f b, v8f c) {
  // V_WMMA_F32_16X16X4_F32 : D = A(16x4 f32) * B(4x16 f32) + C(16x16 f32)
  return __builtin_amdgcn_wmma_f32_16x16x4_f32(false, a, false, b, (short)0, c,
                                               false, false);
}

// ---------------------------------------------------------------------------
// Fused VAE-sample + 9-round residual VQ. One workgroup = (batch b, 16 t's).
// Residual tile (1024 x 16 f32) lives in LDS for in-proj B-fragments, but
// each wave's own 128 rows are ALSO register-resident (z0 - zqacc), so the
// out-proj update is a pure ds_store (no LDS read-modify-write).
// ---------------------------------------------------------------------------
__global__ __launch_bounds__(256) void rvq_fused_kernel(
    const float* __restrict__ x, const float* __restrict__ noise,
    const float* __restrict__ Wpad, const float* __restrict__ b_in,
    const float* __restrict__ cb, const float* __restrict__ W_out,
    const float* __restrict__ b_out, const float* __restrict__ cbnP,
    float* __restrict__ out) {
  __shared__ float Rsh[ND * RS];        // residual tile  (72 KB)
  __shared__ float zepart[8 * 256];     // per-wave in-proj partials (8 KB)
  __shared__ float argsV[TT * 128];     // argmax candidates (8 KB)
  __shared__ int   argsI[TT * 128];     // argmax indices    (8 KB)
  __shared__ float pv[256];             // argmax stage-2 vals
  __shared__ int   pi[256];             // argmax stage-2 idxs
  __shared__ float zel[NDCB * ZS];      // z_e (8 x 16)
  __shared__ float qsh[NDCB * ZS];      // q   (8 x 16)
  __shared__ float red[256];            // scalar reductions

  const int tid = threadIdx.x;
  const int wv  = tid >> 5;             // wave id 0..7
  const int L   = tid & 31;             // lane
  const int lh  = L & 15;               // lane % 16  (N column / t)
  const int hi  = L >> 4;               // half-wave select
  const int bidx = blockIdx.x >> 8;     // batch
  const int t0   = (blockIdx.x & 255) * TT;
  const int koff = hi ? 2 : 0;          // K-row offset per half-wave

  // ---- Phase 0: VAE sample with the WMMA fragment (m,t) mapping ----
  // This thread owns rows  mr = wv*128 + 16j + r + 8*hi  at column t = lh.
  float z0[8][8];                       // z  (register-resident fragment)
  float zqacc[8][8];                    // running z_q for same slots
  float klLocal = 0.f;
  {
    const int rbase = wv * 128 + 8 * hi;
#pragma unroll
    for (int j = 0; j < 8; ++j) {
#pragma unroll
      for (int r = 0; r < 8; ++r) {
        int c = rbase + 16 * j + r;
        float m  = x[((size_t)(bidx * 2 * ND + c)) * NT + t0 + lh];
        float s  = x[((size_t)(bidx * 2 * ND + ND + c)) * NT + t0 + lh];
        float nz = noise[((size_t)(bidx * ND + c)) * NT + t0 + lh];
        float sp = (s > 20.f) ? s : log1pf(__expf(s));  // softplus
        float st = sp + 1e-4f;
        float var = st * st;
        float lv  = 2.f * __logf(st);
        klLocal += m * m + var - lv - 1.f;
        float z = nz * st + m;
        z0[j][r]    = z;
        zqacc[j][r] = 0.f;
        Rsh[c * RS + lh] = z;
      }
    }
  }
  float commitLocal = 0.f;
  __syncthreads();

  const v2f* cbp = reinterpret_cast<const v2f*>(cbnP);

  for (int i = 0; i < NCB; ++i) {
    __builtin_prefetch(cbnP + (size_t)i * NDCB * NK, 0, 3);
    __builtin_prefetch(cb + (size_t)i * NK * NDCB, 0, 3);

    // ---- in-proj: z_e(8x16) = W_in[i](8x1024) @ R(1024x16) + b_in ----
    {
      // per-lane padded row pointer: rows 8..15 are zeros -> no divergence
      const float* wrow = Wpad + ((size_t)(i * 16) + lh) * ND;
      v8f acc;
#pragma unroll
      for (int r = 0; r < 8; ++r) acc[r] = 0.f;
      const int kbase = wv * 128 + koff; // each wave owns 128 of K=1024
#pragma unroll 4
      for (int ch = 0; ch < 32; ++ch) {
        int k0 = kbase + ch * 4;
        v2f a = *reinterpret_cast<const v2f*>(wrow + k0);        // b64
        v2f b; b.x = Rsh[k0 * RS + lh]; b.y = Rsh[(k0 + 1) * RS + lh];
        acc = wmma_f32(a, b, acc);
      }
#pragma unroll
      for (int r = 0; r < 8; ++r)
        zepart[wv * 256 + (r + hi * 8) * 16 + lh] = acc[r];
    }
    __syncthreads();
    {
      int ml = tid >> 4;                // tid == ml*16 + t
      if (ml < NDCB) {
        float s = 0.f;
#pragma unroll
        for (int w = 0; w < 8; ++w) s += zepart[w * 256 + tid];
        zel[ml * ZS + (tid & 15)] = s + b_in[i * NDCB + ml];
      }
    }
    __syncthreads();

    // ---- search: sim(16t x 1024k) = z_e^T @ cb_n^T ; running argmax ----
    float vmax[8]; int vidx[8];
#pragma unroll
    for (int r = 0; r < 8; ++r) { vmax[r] = -3.0e38f; vidx[r] = 0; }
    v2f a0, a1;                          // A = z_e^T (16 x 8), load once
    a0.x = zel[(0 + koff) * ZS + lh]; a0.y = zel[(1 + koff) * ZS + lh];
    a1.x = zel[(4 + koff) * ZS + lh]; a1.y = zel[(5 + koff) * ZS + lh];
    const int p0 = (i * 4 + hi) * NK;        // pair index for rows koff,koff+1
    const int p1 = (i * 4 + 2 + hi) * NK;    // pair index for rows 4+koff,...
#pragma unroll
    for (int j = 0; j < 8; ++j) {        // 8 N-tiles of 16 codes per wave
      int nb = (wv * 8 + j) * 16;
      v2f b0 = cbp[p0 + nb + lh];        // single global_load_b64 each
      v2f b1 = cbp[p1 + nb + lh];
      v8f cc;
#pragma unroll
      for (int r = 0; r < 8; ++r) cc[r] = 0.f;
      cc = wmma_f32(a0, b0, cc);
      cc = wmma_f32(a1, b1, cc);
#pragma unroll
      for (int r = 0; r < 8; ++r)
        if (cc[r] > vmax[r]) { vmax[r] = cc[r]; vidx[r] = nb + lh; }
    }
#pragma unroll
    for (int r = 0; r < 8; ++r) {
      int t = r + hi * 8;                // C row M = t
      argsV[t * 128 + wv * 16 + lh] = vmax[r];
      argsI[t * 128 + wv * 16 + lh] = vidx[r];
    }
    __syncthreads();
    {                                    // argmax stage 1: 256 threads x 8
      int t = tid >> 4, g = tid & 15;
      float bv = -3.0e38f; int bi = 0;
#pragma unroll
      for (int n = 0; n < 8; ++n) {
        float v = argsV[t * 128 + g * 8 + n];
        if (v > bv) { bv = v; bi = argsI[t * 128 + g * 8 + n]; }
      }
      pv[tid] = bv; pi[tid] = bi;
    }
    __syncthreads();
    if (tid < TT) {                      // stage 2 + gather q + commit
      float bv = -3.0e38f; int bi = 0;
#pragma unroll
      for (int g = 0; g < 16; ++g) {
        float v = pv[tid * 16 + g];
        if (v > bv) { bv = v; bi = pi[tid * 16 + g]; }
      }
      const float* q = cb + ((size_t)i * NK + bi) * NDCB;
      float cl = 0.f;
#pragma unroll
      for (int d = 0; d < NDCB; ++d) {
        float qd = q[d];
        qsh[d * ZS + tid] = qd;
        float diff = zel[d * ZS + tid] - qd;
        cl += diff * diff;
      }
      commitLocal += cl;
    }
    __syncthreads();

    // ---- out-proj: z_q_i = W_out[i](1024x8) @ q(8x16) + b_out ----
    // Residual update is a PURE ds_store: R_new = z0 - zqacc_new.
    {
      v2f bq0, bq1;                      // B = q, load once
      bq0.x = qsh[(0 + koff) * ZS + lh]; bq0.y = qsh[(1 + koff) * ZS + lh];
      bq1.x = qsh[(4 + koff) * ZS + lh]; bq1.y = qsh[(5 + koff) * ZS + lh];
      const bool storeR = (i != NCB - 1);
#pragma unroll
      for (int j = 0; j < 8; ++j) {      // 8 M-tiles of 16 rows per wave
        int mt = (wv * 8 + j) * 16;
        const float* wr = W_out + ((size_t)i * ND + mt + lh) * NDCB;
        v2f aa0 = *reinterpret_cast<const v2f*>(wr + koff);      // b64
        v2f aa1 = *reinterpret_cast<const v2f*>(wr + 4 + koff);  // b64
        v8f cc;
#pragma unroll
        for (int r = 0; r < 8; ++r) cc[r] = 0.f;
        cc = wmma_f32(aa0, bq0, cc);
        cc = wmma_f32(aa1, bq1, cc);
#pragma unroll
        for (int r = 0; r < 8; ++r) {
          int mr = mt + r + hi * 8;
          float val = cc[r] + b_out[i * ND + mr];
          float zq  = zqacc[j][r] + val;
          zqacc[j][r] = zq;
          if (storeR) Rsh[mr * RS + lh] = z0[j][r] - zq;  // residual
        }
      }
    }
    __syncthreads();
  }

  // ---- write z_q ----
#pragma unroll
  for (int j = 0; j < 8; ++j)
#pragma unroll
    for (int r = 0; r < 8; ++r) {
      int mr = (wv * 8 + j) * 16 + r + hi * 8;
      out[((size_t)(bidx * ND + mr)) * NT + t0 + lh] = zqacc[j][r];
    }

  // ---- scalar reductions: kl, commit (== cbloss forward) ----
  const size_t BDT = (size_t)NBAT * ND * NT;
  red[tid] = klLocal;
  __syncthreads();
  if (tid == 0) {
    float s = 0.f;
    for (int n = 0; n < 256; ++n) s += red[n];
    atomicAdd(out + BDT, s * (1.f / 32768.f));       // mean over B*T
  }
  __syncthreads();
  red[tid] = commitLocal;
  __syncthreads();
  if (tid == 0) {
    float s = 0.f;
    for (int n = 0; n < 256; ++n) s += red[n];
    float v = s * (1.f / 2359296.f);                 // /(9 * B*Dcb*T)
    atomicAdd(out + BDT + 1, v);
    atomicAdd(out + BDT + 2, v);
  }
}

extern "C" void kernel_launch(void* const* d_in, const int* in_sizes, int n_in,
                              void* d_out, int out_size, void* d_ws, size_t ws_size,
                              hipStream_t stream) {
  const float* x         = (const float*)d_in[0];
  const float* noise     = (const float*)d_in[1];
  const float* W_in      = (const float*)d_in[2];
  const float* b_in      = (const float*)d_in[3];
  const float* codebooks = (const float*)d_in[4];
  const float* W_out     = (const float*)d_in[5];
  const float* b_out     = (const float*)d_in[6];
  float* out  = (float*)d_out;
  float* cbnP = (float*)d_ws;                                   // 288 KB
  float* Wpad = (float*)d_ws + (size_t)NCB * NDCB * NK * 2;     // 576 KB

  const size_t BDT = (size_t)NBAT * ND * NT;
  hipMemsetAsync(out + BDT, 0, 3 * sizeof(float), stream);  // zero scalar slots

  cbnorm_kernel<<<(NCB * NK + 255) / 256, 256, 0, stream>>>(codebooks, cbnP);
  wpad_kernel<<<(NCB * 16 * ND + 255) / 256, 256, 0, stream>>>(W_in, Wpad);

  // grid: 8 batches x 256 time-tiles = 2048 workgroups, 256 threads (8 waves)
  rvq_fused_kernel<<<NBAT * (NT / TT), 256, 0, stream>>>(
      x, noise, Wpad, b_in, codebooks, W_out, b_out, cbnP, out);
}